// DeepIRT_5033701671028
// MI455X (gfx1250) — compile-verified
//
#include <hip/hip_runtime.h>
#include <hip/hip_bf16.h>
#include <math.h>

// ---------------------------------------------------------------------------
// DeepIRT on MI455X (gfx1250, wave32).
//   B=64, L=2048, D=64, M=50, NUM_Q=10000.
// Kernel 1: gathers + (k@Mk^T -> softmax w), (v@We -> sigmoid e), (v@Wa -> tanh a)
//           using fp32 WMMA V_WMMA_F32_16X16X4_F32 (16 rows per wave).
//           Mk is staged zero-padded (50->64 rows) in LDS so the padded GEMM
//           inner loop is branchless (no exec-mask churn).
// Kernel 2: sequential scan over L; one block per batch, Mv column in VGPRs,
//           w double-buffered in LDS, one barrier per timestep.
// Kernel 3: f = tanh([read|k] @ Wf + bf) via fp32 WMMA, then per-row dots with
//           Wab / Wd and the logit combine.
// ---------------------------------------------------------------------------

#define BB     64
#define LL     2048
#define DD     64
#define MM     50
#define NQ     10000
#define BL     (BB * LL)          // 131072 rows

typedef float v2f __attribute__((ext_vector_type(2)));
typedef float v8f __attribute__((ext_vector_type(8)));

static __device__ inline v8f wmma_f32_k4(v2f a, v2f b, v8f c) {
    // D = A(16x4,f32) * B(4x16,f32) + C(16x16,f32)
    return __builtin_amdgcn_wmma_f32_16x16x4_f32(false, a, false, b, (short)0, c,
                                                 false, false);
}

static __device__ inline float sigmoidf_(float x) { return 1.0f / (1.0f + expf(-x)); }

// ---------------------------------------------------------------------------
// Kernel 1: per-row front end. 4 waves/block, one 16-row tile per wave.
// ---------------------------------------------------------------------------
__global__ __launch_bounds__(128)
void deepirt_frontend(const int* __restrict__ question,
                      const int* __restrict__ response,
                      const float* __restrict__ k_emb,
                      const float* __restrict__ v_emb,
                      const float* __restrict__ Mk,
                      const float* __restrict__ We, const float* __restrict__ be,
                      const float* __restrict__ Wa, const float* __restrict__ ba,
                      float* __restrict__ ws_w,    // [BL][50]
                      float* __restrict__ ws_e,    // [BL][64]
                      float* __restrict__ ws_a)    // [BL][64]
{
    __shared__ float lds_k[4][16][68];   // padded stride 68 -> conflict-free
    __shared__ float lds_v[4][16][68];
    __shared__ float lds_wl[4][16][68];  // w-logit dump for softmax
    __shared__ float lds_mk[64][68];     // zero-padded Mk (block-shared)
    __shared__ int   lds_q[4][16];
    __shared__ int   lds_x[4][16];

    const int tid  = threadIdx.x;
    const int w    = tid >> 5;
    const int lane = tid & 31;
    const int tile = blockIdx.x * 4 + w;
    const int row0 = tile * 16;

    if (lane < 16) {
        int r = row0 + lane;
        int q = question[r];
        lds_q[w][lane] = q;
        lds_x[w][lane] = q + NQ * response[r];
    }
    // stage zero-padded Mk (64x64 logical) once per block, coalesced
    for (int idx = tid; idx < 64 * 64; idx += 128) {
        int n = idx >> 6, c = idx & 63;
        lds_mk[n][c] = (n < MM) ? Mk[n * DD + c] : 0.0f;
    }
    __syncthreads();

    // gather k / v tiles (16x64 each), 32 elems per lane
    for (int idx = lane; idx < 16 * 64; idx += 32) {
        int r = idx >> 6, c = idx & 63;
        lds_k[w][r][c] = k_emb[(size_t)lds_q[w][r] * DD + c];
        lds_v[w][r][c] = v_emb[(size_t)lds_x[w][r] * DD + c];
    }
    __syncthreads();

    const int rA = lane & 15;        // A-matrix row held by this lane
    const int nA = lane & 15;        // B-matrix column held by this lane
    const int kb = (lane >> 4) * 2;  // K sub-offset (lanes 16-31 take K+2,K+3)

    // -------- w logits: k(16x64) @ Mk^T(64x50, zero-padded to 64) --------
    v8f accw[4];
    v8f zero8 = {0, 0, 0, 0, 0, 0, 0, 0};
#pragma unroll
    for (int nt = 0; nt < 4; nt++) accw[nt] = zero8;

#pragma unroll 4
    for (int kk = 0; kk < 16; kk++) {
        const int kr = 4 * kk + kb;
        v2f A; A.x = lds_k[w][rA][kr]; A.y = lds_k[w][rA][kr + 1];
#pragma unroll
        for (int nt = 0; nt < 4; nt++) {
            const int n = nA + nt * 16;
            v2f Bf; Bf.x = lds_mk[n][kr]; Bf.y = lds_mk[n][kr + 1];
            accw[nt] = wmma_f32_k4(A, Bf, accw[nt]);
        }
    }
    // dump logits -> LDS and softmax (lanes 0-15, one row each)
#pragma unroll
    for (int nt = 0; nt < 4; nt++)
#pragma unroll
        for (int i = 0; i < 8; i++)
            lds_wl[w][i + (lane >> 4) * 8][nA + nt * 16] = accw[nt][i];
    __syncthreads();

    if (lane < 16) {
        const int r = row0 + lane;
        float mx = -1e30f;
        for (int m = 0; m < MM; m++) mx = fmaxf(mx, lds_wl[w][lane][m]);
        float s = 0.0f;
        for (int m = 0; m < MM; m++) {
            float ev = expf(lds_wl[w][lane][m] - mx);
            lds_wl[w][lane][m] = ev;
            s += ev;
        }
        const float inv = 1.0f / s;
        for (int m = 0; m < MM; m++)
            ws_w[(size_t)r * MM + m] = lds_wl[w][lane][m] * inv;
    }

    // -------- e = sigmoid(v@We+be), a = tanh(v@Wa+ba) --------
    v8f acce[4], acca[4];
#pragma unroll
    for (int nt = 0; nt < 4; nt++) { acce[nt] = zero8; acca[nt] = zero8; }

#pragma unroll 4
    for (int kk = 0; kk < 16; kk++) {
        const int kr = 4 * kk + kb;
        v2f A; A.x = lds_v[w][rA][kr]; A.y = lds_v[w][rA][kr + 1];
#pragma unroll
        for (int nt = 0; nt < 4; nt++) {
            const int n = nA + nt * 16;
            v2f Be; Be.x = We[kr * DD + n]; Be.y = We[(kr + 1) * DD + n];
            acce[nt] = wmma_f32_k4(A, Be, acce[nt]);
            v2f Ba; Ba.x = Wa[kr * DD + n]; Ba.y = Wa[(kr + 1) * DD + n];
            acca[nt] = wmma_f32_k4(A, Ba, acca[nt]);
        }
    }
#pragma unroll
    for (int nt = 0; nt < 4; nt++) {
        const int n   = nA + nt * 16;
        const float beN = be[n];
        const float baN = ba[n];
#pragma unroll
        for (int i = 0; i < 8; i++) {
            const int r = row0 + i + (lane >> 4) * 8;
            ws_e[(size_t)r * DD + n] = sigmoidf_(acce[nt][i] + beN);
            ws_a[(size_t)r * DD + n] = tanhf(acca[nt][i] + baN);
        }
    }
}

// ---------------------------------------------------------------------------
// Kernel 2: sequential DKVMN scan. One block per batch (64 blocks x 64 threads).
// Thread d keeps column d of Mv (50 floats) in registers.
// ---------------------------------------------------------------------------
__global__ __launch_bounds__(64)
void deepirt_scan(const float* __restrict__ ws_w,
                  const float* __restrict__ ws_e,
                  const float* __restrict__ ws_a,
                  const float* __restrict__ Mv0,
                  float* __restrict__ ws_r)
{
    const int b = blockIdx.x;
    const int d = threadIdx.x;

    __shared__ float wsh[2][MM];

    float Mv[MM];
#pragma unroll
    for (int m = 0; m < MM; m++) Mv[m] = Mv0[m * DD + d];

    if (d < MM) wsh[0][d] = ws_w[(size_t)b * LL * MM + d];
    __syncthreads();

    for (int t = 0; t < LL; t++) {
        const size_t row = (size_t)b * LL + t;

        // prefetch next step's w while computing this step
        float wnext = 0.0f;
        if ((t + 1) < LL && d < MM) wnext = ws_w[(row + 1) * MM + d];

        const float e = ws_e[row * DD + d];
        const float a = ws_a[row * DD + d];

        const float* wb = wsh[t & 1];
        float rd = 0.0f;
#pragma unroll
        for (int m = 0; m < MM; m++) {
            const float wm = wb[m];
            const float mv = Mv[m];
            rd += wm * mv;                       // read BEFORE the update
            Mv[m] = mv - wm * e * mv + wm * a;   // mv*(1 - wm*e) + wm*a
        }
        ws_r[row * DD + d] = rd;

        if ((t + 1) < LL && d < MM) wsh[(t + 1) & 1][d] = wnext;
        __syncthreads();
    }
}

// ---------------------------------------------------------------------------
// Kernel 3: f = tanh([read|k] @ Wf + bf); logits = 3*tanh(f.Wab+bab) - tanh(k.Wd+bd)
// ---------------------------------------------------------------------------
__global__ __launch_bounds__(128)
void deepirt_backend(const int* __restrict__ question,
                     const float* __restrict__ k_emb,
                     const float* __restrict__ ws_r,
                     const float* __restrict__ Wf, const float* __restrict__ bf,
                     const float* __restrict__ Wab, const float* __restrict__ bab,
                     const float* __restrict__ Wd,  const float* __restrict__ bd,
                     float* __restrict__ out)
{
    __shared__ float lds_in[4][16][132];  // cols 0-63: read, 64-127: k (stride 132)
    __shared__ float lds_f[4][16][68];
    __shared__ int   lds_q[4][16];

    const int w    = threadIdx.x >> 5;
    const int lane = threadIdx.x & 31;
    const int tile = blockIdx.x * 4 + w;
    const int row0 = tile * 16;

    if (lane < 16) lds_q[w][lane] = question[row0 + lane];
    __syncthreads();

    for (int idx = lane; idx < 16 * 128; idx += 32) {
        int r = idx >> 7, c = idx & 127;
        float val = (c < 64) ? ws_r[(size_t)(row0 + r) * DD + c]
                             : k_emb[(size_t)lds_q[w][r] * DD + (c - 64)];
        lds_in[w][r][c] = val;
    }
    __syncthreads();

    const int rA = lane & 15;
    const int nA = lane & 15;
    const int kb = (lane >> 4) * 2;

    v8f accf[4];
    v8f zero8 = {0, 0, 0, 0, 0, 0, 0, 0};
#pragma unroll
    for (int nt = 0; nt < 4; nt++) accf[nt] = zero8;

#pragma unroll 4
    for (int kk = 0; kk < 32; kk++) {            // K = 128
        const int kr = 4 * kk + kb;
        v2f A; A.x = lds_in[w][rA][kr]; A.y = lds_in[w][rA][kr + 1];
#pragma unroll
        for (int nt = 0; nt < 4; nt++) {
            const int n = nA + nt * 16;
            v2f Bf; Bf.x = Wf[kr * DD + n]; Bf.y = Wf[(kr + 1) * DD + n];
            accf[nt] = wmma_f32_k4(A, Bf, accf[nt]);
        }
    }
#pragma unroll
    for (int nt = 0; nt < 4; nt++) {
        const int n   = nA + nt * 16;
        const float bfn = bf[n];
#pragma unroll
        for (int i = 0; i < 8; i++)
            lds_f[w][i + (lane >> 4) * 8][n] = tanhf(accf[nt][i] + bfn);
    }
    __syncthreads();

    if (lane < 16) {
        const int r = row0 + lane;
        float sA = 0.0f, sD = 0.0f;
        for (int c = 0; c < DD; c++) {
            sA += lds_f[w][lane][c] * Wab[c];
            sD += lds_in[w][lane][64 + c] * Wd[c];
        }
        const float ab = tanhf(sA + bab[0]);
        const float qd = tanhf(sD + bd[0]);
        out[r] = 3.0f * ab - qd;
    }
}

// ---------------------------------------------------------------------------
extern "C" void kernel_launch(void* const* d_in, const int* in_sizes, int n_in,
                              void* d_out, int out_size, void* d_ws, size_t ws_size,
                              hipStream_t stream) {
    (void)in_sizes; (void)n_in; (void)out_size; (void)ws_size;

    const int*   question = (const int*)d_in[0];
    const int*   response = (const int*)d_in[1];
    const float* k_emb    = (const float*)d_in[2];
    const float* v_emb    = (const float*)d_in[3];
    const float* Mk       = (const float*)d_in[4];
    const float* Mv0      = (const float*)d_in[5];
    const float* Wf       = (const float*)d_in[6];
    const float* bf       = (const float*)d_in[7];
    const float* We       = (const float*)d_in[8];
    const float* be       = (const float*)d_in[9];
    const float* Wa       = (const float*)d_in[10];
    const float* ba       = (const float*)d_in[11];
    const float* Wab      = (const float*)d_in[12];
    const float* bab      = (const float*)d_in[13];
    const float* Wd       = (const float*)d_in[14];
    const float* bd       = (const float*)d_in[15];
    float* out = (float*)d_out;

    // workspace layout (floats): w[BL*50] | e[BL*64] | a[BL*64] | read[BL*64]
    float* ws   = (float*)d_ws;
    float* ws_w = ws;
    float* ws_e = ws_w + (size_t)BL * MM;
    float* ws_a = ws_e + (size_t)BL * DD;
    float* ws_r = ws_a + (size_t)BL * DD;

    const int tiles  = BL / 16;      // 8192
    const int blocks = tiles / 4;    // 2048 (4 waves/block)

    deepirt_frontend<<<blocks, 128, 0, stream>>>(question, response, k_emb, v_emb,
                                                 Mk, We, be, Wa, ba,
                                                 ws_w, ws_e, ws_a);
    deepirt_scan<<<BB, 64, 0, stream>>>(ws_w, ws_e, ws_a, Mv0, ws_r);
    deepirt_backend<<<blocks, 128, 0, stream>>>(question, k_emb, ws_r,
                                                Wf, bf, Wab, bab, Wd, bd, out);
}